// ClassCaps_57638461112778
// MI455X (gfx1250) — compile-verified
//
#include <hip/hip_runtime.h>
#include <math.h>

#define B_    32
#define HW_   144
#define IN_   32
#define OUT_  10
#define NIN_  4608      // HW_ * IN_
#define PL_   16
#define EPSF  1e-8f

typedef __attribute__((ext_vector_type(2))) float v2f;
typedef __attribute__((ext_vector_type(8))) float v8f;

// ---------------------------------------------------------------------------
// Kernel 1: vote transform via V_WMMA_F32_16X16X4_F32.
// Per input capsule i: GEMM  A[18432 x 4] @ B[4 x 40] -> D[18432 x 40]
//   rows  = (b*144 + n_pix)*4 + p       (pose row p)
//   cols  = j*4 + r                      (out capsule j, pose col r)
//   K = q = 4  -> exactly the 16x16x4 f32 WMMA shape.
// One wave = one 16-row M-tile for one i; 3 N-tiles (40 cols, zero-padded).
// ---------------------------------------------------------------------------
__global__ __launch_bounds__(256)
void votes_wmma_kernel(const float* __restrict__ caps,   // [B,12,12,IN,17]
                       const float* __restrict__ trans,  // [IN,OUT,4,4]
                       float* __restrict__ V)            // [B,NIN,OUT,16]
{
    const int lane  = threadIdx.x & 31;
    const int wave  = blockIdx.x * (blockDim.x >> 5) + (threadIdx.x >> 5);
    const int i     = wave / 1152;           // input capsule
    const int mtile = wave - i * 1152;       // 18432/16 = 1152 row tiles
    const int half  = lane >> 4;             // K-half select
    const int m     = lane & 15;

    // ---- A fragment (16x4 f32): lanes 0-15 -> K=0,1 ; lanes 16-31 -> K=2,3
    const int row = mtile * 16 + m;          // (b*144+n_pix)*4 + p
    const int p   = row & 3;
    const int bn  = row >> 2;                // b*144 + n_pix
    const float* ap = caps + (size_t)(bn * IN_ + i) * 17 + p * 4 + 2 * half;
    v2f A; A.x = ap[0]; A.y = ap[1];

    for (int jt = 0; jt < 3; ++jt) {
        // ---- B fragment (4x16 f32): same K-half split, col = m
        v2f Bf; Bf.x = 0.0f; Bf.y = 0.0f;
        const int c = jt * 16 + m;           // global column (j*4+r), pad >=40
        if (c < 40) {
            const int j = c >> 2, r = c & 3;
            const float* bp = trans + ((size_t)(i * OUT_ + j) * 4 + 2 * half) * 4 + r;
            Bf.x = bp[0];                    // q = 2*half
            Bf.y = bp[4];                    // q = 2*half + 1
        }
        v8f C = {0.f,0.f,0.f,0.f,0.f,0.f,0.f,0.f};
        v8f D = __builtin_amdgcn_wmma_f32_16x16x4_f32(
            /*neg_a=*/false, A, /*neg_b=*/false, Bf,
            /*c_mod=*/(short)0, C, /*reuse_a=*/false, /*reuse_b=*/false);

        // ---- store: lane holds column c, VGPR v -> row (v + 8*half)
        if (c < 40) {
            const int j = c >> 2, r = c & 3;
            #pragma unroll
            for (int v = 0; v < 8; ++v) {
                const int rowd = mtile * 16 + (v + 8 * half);
                const int pd   = rowd & 3;
                const int bnd  = rowd >> 2;      // b*144 + n_pix
                float val = D[v];
                const int P = pd * 4 + r;        // flattened pose index
                if (P < 2) {                     // coordinate addition
                    const int n_pix = bnd % HW_;
                    val += (P == 0) ? (float)(n_pix / 12) * (1.0f / 12.0f)
                                    : (float)(n_pix % 12) * (1.0f / 12.0f);
                }
                V[((size_t)(bnd * IN_ + i) * OUT_ + j) * PL_ + P] = val;
            }
        }
    }
}

// ---------------------------------------------------------------------------
// Kernel 2: EM routing — one workgroup per batch element (V stays in L2).
// ---------------------------------------------------------------------------
__global__ __launch_bounds__(512)
void em_routing_kernel(const float* __restrict__ caps,
                       const float* __restrict__ beta_u_p,
                       const float* __restrict__ beta_a_p,
                       const float* __restrict__ V,   // [B,NIN,OUT,16]
                       float* __restrict__ R,         // [B,NIN,OUT] scratch
                       float* __restrict__ out)       // [B,OUT,17]
{
    const int b   = blockIdx.x;
    const int tid = threadIdx.x;
    const float beta_u = beta_u_p[0];
    const float beta_a = beta_a_p[0];
    const float* Vb = V + (size_t)b * NIN_ * OUT_ * PL_;
    float*       Rb = R + (size_t)b * NIN_ * OUT_;

    __shared__ float sA[NIN_];
    __shared__ float sS1[OUT_][PL_];
    __shared__ float sS2[OUT_][PL_];
    __shared__ float sRsum[OUT_];
    __shared__ float sMu[OUT_][PL_];
    __shared__ float sInvSig[OUT_][PL_];
    __shared__ float sLt[OUT_][PL_];    // log(2*pi*sigma2)
    __shared__ float sCost[OUT_][PL_];
    __shared__ float sAout[OUT_];
    __shared__ float sLogA[OUT_];

    for (int n = tid; n < NIN_; n += 512)
        sA[n] = caps[(size_t)(b * NIN_ + n) * 17 + 16];
    for (int k = tid; k < NIN_ * OUT_; k += 512)
        Rb[k] = 1.0f / (float)OUT_;
    __syncthreads();

    const int p = tid & 15;     // pose component handled by this thread
    const int g = tid >> 4;     // n-group 0..31

    for (int t = 0; t < 3; ++t) {
        const float lam = 1.0f + 2.0f * (float)t;   // 1, 3, 5

        if (tid < OUT_ * PL_) { sS1[tid >> 4][tid & 15] = 0.0f;
                                sS2[tid >> 4][tid & 15] = 0.0f; }
        if (tid < OUT_) sRsum[tid] = 0.0f;
        __syncthreads();

        // ---- M-step accumulation: S0 = sum rw, S1 = sum rw*V, S2 = sum rw*V^2
        float a0[OUT_], a1[OUT_], a2[OUT_];
        #pragma unroll
        for (int j = 0; j < OUT_; ++j) { a0[j] = a1[j] = a2[j] = 0.0f; }
        for (int n = g; n < NIN_; n += 32) {
            const float a = sA[n];
            const float* vrow = Vb + (size_t)n * (OUT_ * PL_);
            const float* rrow = Rb + (size_t)n * OUT_;
            #pragma unroll
            for (int j = 0; j < OUT_; ++j) {
                const float rw = rrow[j] * a;
                const float v  = vrow[j * PL_ + p];
                a1[j] += rw * v;
                a2[j] += rw * v * v;
                a0[j] += rw;
            }
        }
        #pragma unroll
        for (int j = 0; j < OUT_; ++j) {
            atomicAdd(&sS1[j][p], a1[j]);
            atomicAdd(&sS2[j][p], a2[j]);
            if (p == 0) atomicAdd(&sRsum[j], a0[j]);
        }
        __syncthreads();

        // ---- mu / sigma2 / cost  (sum rw (V-mu)^2 = S2 - 2 mu S1 + mu^2 S0)
        if (tid < OUT_ * PL_) {
            const int j = tid >> 4, q = tid & 15;
            const float s0 = sRsum[j];
            const float rs = s0 + EPSF;
            const float s1 = sS1[j][q];
            const float mu = s1 / rs;
            const float sig = (sS2[j][q] - 2.0f * mu * s1 + mu * mu * s0) / rs + EPSF;
            sMu[j][q]     = mu;
            sInvSig[j][q] = 1.0f / sig;
            sLt[j][q]     = __logf(6.283185307179586f * sig);
            sCost[j][q]   = (beta_u + 0.5f * __logf(sig)) * rs;
        }
        __syncthreads();
        if (tid < OUT_) {
            float cs = 0.0f;
            #pragma unroll
            for (int q = 0; q < PL_; ++q) cs += sCost[tid][q];
            const float ao = 1.0f / (1.0f + __expf(-(lam * (beta_a - cs))));
            sAout[tid] = ao;
            sLogA[tid] = __logf(ao + EPSF);
        }
        __syncthreads();

        // ---- E-step (skipped after last iteration)
        if (t < 2) {
            for (int n = tid; n < NIN_; n += 512) {
                const float* vrow = Vb + (size_t)n * (OUT_ * PL_);
                float lg[OUT_];
                float mx = -3.4e38f;
                #pragma unroll
                for (int j = 0; j < OUT_; ++j) {
                    float lp = 0.0f;
                    #pragma unroll
                    for (int q = 0; q < PL_; ++q) {
                        const float d = vrow[j * PL_ + q] - sMu[j][q];
                        lp += d * d * sInvSig[j][q] + sLt[j][q];
                    }
                    const float l = sLogA[j] - 0.5f * lp;
                    lg[j] = l;
                    mx = fmaxf(mx, l);
                }
                float se = 0.0f;
                #pragma unroll
                for (int j = 0; j < OUT_; ++j) {
                    const float e = __expf(lg[j] - mx);
                    lg[j] = e; se += e;
                }
                const float inv = 1.0f / se;
                float* rrow = Rb + (size_t)n * OUT_;
                #pragma unroll
                for (int j = 0; j < OUT_; ++j) rrow[j] = lg[j] * inv;
            }
            __syncthreads();
        }
    }

    // ---- output: [B, OUT, 17] = concat(mu, a_out)
    if (tid < OUT_ * PL_)
        out[(size_t)(b * OUT_ + (tid >> 4)) * 17 + (tid & 15)] = sMu[tid >> 4][tid & 15];
    if (tid < OUT_)
        out[(size_t)(b * OUT_ + tid) * 17 + 16] = sAout[tid];
}

// ---------------------------------------------------------------------------
extern "C" void kernel_launch(void* const* d_in, const int* in_sizes, int n_in,
                              void* d_out, int out_size, void* d_ws, size_t ws_size,
                              hipStream_t stream)
{
    (void)in_sizes; (void)n_in; (void)out_size; (void)ws_size;
    const float* caps  = (const float*)d_in[0];   // [32,12,12,32,17]
    const float* trans = (const float*)d_in[1];   // [1,1,32,10,4,4]
    const float* bu    = (const float*)d_in[2];
    const float* ba    = (const float*)d_in[3];
    float* out = (float*)d_out;

    float* V = (float*)d_ws;                                // 94,371,840 B
    float* R = V + (size_t)B_ * NIN_ * OUT_ * PL_;          // + 5,898,240 B

    // 32 capsules * 1152 M-tiles = 36864 waves; 8 waves/block -> 4608 blocks
    votes_wmma_kernel<<<4608, 256, 0, stream>>>(caps, trans, V);
    em_routing_kernel<<<B_, 512, 0, stream>>>(caps, bu, ba, V, R, out);
}